// ParConv2d_50233937494204
// MI455X (gfx1250) — compile-verified
//
#include <hip/hip_runtime.h>

// Per-batch depthwise 3x3 conv, pad=1, stride=1.
// x: (B=16, C=64, H=256, W=256) f32; kernel: (B, 3, 3) f32; out same shape as x.
// Memory-bound (AI ~2.25 F/B, ~23us floor at 23.3 TB/s): pure-bandwidth design.
// CDNA5 path: async global->LDS staging (ASYNCcnt) + branch-free LDS stencil.

#define BB 16
#define CC 64
#define HH 256
#define WW 256
#define TILE_H 32
#define TILES_PER_PLANE (HH / TILE_H)   // 8
#define LDS_ROWS (TILE_H + 2)           // 34 rows: halo + tile + halo
#define LDS_W 264                       // cols 0..2 pad, col 3 = left-zero, 4..259 data (16B aligned), 260 = right-zero

typedef int v4i __attribute__((ext_vector_type(4)));
typedef __attribute__((address_space(1))) v4i* gptr_v4i;
typedef __attribute__((address_space(3))) v4i* lptr_v4i;

__device__ __forceinline__ void async_copy16(const float* gp, float* lp) {
#if defined(__gfx1250__) && __has_builtin(__builtin_amdgcn_global_load_async_to_lds_b128)
    __builtin_amdgcn_global_load_async_to_lds_b128(
        (gptr_v4i)(v4i*)(void*)const_cast<float*>(gp),
        (lptr_v4i)(v4i*)(void*)lp,
        /*offset=*/0, /*cpol=*/0);
#else
    *(float4*)lp = *(const float4*)gp;
#endif
}

__device__ __forceinline__ void wait_async_lds() {
#if defined(__gfx1250__) && __has_builtin(__builtin_amdgcn_s_wait_asynccnt)
    __builtin_amdgcn_s_wait_asynccnt(0);
#elif defined(__gfx1250__)
    asm volatile("s_wait_asynccnt 0" ::: "memory");
#endif
}

__global__ __launch_bounds__(256) void dwconv3x3_kernel(const float* __restrict__ x,
                                                        const float* __restrict__ kern,
                                                        float* __restrict__ out) {
    __shared__ float tile[LDS_ROWS][LDS_W];

    const int tid   = threadIdx.x;              // 0..255 = output column
    const int blk   = blockIdx.x;               // 8192 blocks
    const int trow  = blk & (TILES_PER_PLANE - 1);
    const int plane = blk >> 3;                 // 0..1023 = b*C + c
    const int bIdx  = plane >> 6;               // /C
    const size_t planeOff = (size_t)plane * (HH * WW);
    const int h0 = trow * TILE_H;

    // 3x3 weights: block-uniform index -> scalar (s_load) path
    const float* kp = kern + bIdx * 9;
    const float k0 = kp[0], k1 = kp[1], k2 = kp[2];
    const float k3 = kp[3], k4 = kp[4], k5 = kp[5];
    const float k6 = kp[6], k7 = kp[7], k8 = kp[8];

    // ---- Bulk staging: interior rows 1..32 (global rows h0..h0+31) are ALWAYS
    // in range. 32 rows * 64 float4 = 2048 copies = exactly 8 per thread,
    // fully unrolled and branch-free -> 8 back-to-back async b128 issues.
    const float* gbase = x + planeOff + (size_t)h0 * WW;
#pragma unroll
    for (int i = 0; i < 8; ++i) {
        const int idx = tid + i * 256;          // 0..2047
        const int rr  = idx >> 6;               // interior row 0..31
        const int q   = idx & 63;               // float4 within row
        async_copy16(gbase + (size_t)rr * WW + q * 4, &tile[rr + 1][4 + q * 4]);
    }

    // ---- Halo rows 0 (h0-1) and 33 (h0+32): 128 lanes, bounds-checked.
    if (tid < 128) {
        const bool top = (tid < 64);
        const int  r   = top ? 0 : (LDS_ROWS - 1);
        const int  q   = tid & 63;
        const int  h   = top ? (h0 - 1) : (h0 + TILE_H);
        float* lp = &tile[r][4 + q * 4];
        if (h >= 0 && h < HH) {
            async_copy16(x + planeOff + (size_t)h * WW + q * 4, lp);
        } else {
            *(float4*)lp = make_float4(0.f, 0.f, 0.f, 0.f);
        }
    } else if (tid < 128 + LDS_ROWS) {
        // Horizontal zero-halo columns -> branch-free compute
        const int r = tid - 128;
        tile[r][3]   = 0.f;
        tile[r][260] = 0.f;
    }

    wait_async_lds();       // drain this wave's ASYNCcnt ...
    __syncthreads();        // ... barrier makes all waves' LDS writes visible

    // ---- Compute: one thread per column, 32 rows, rolling 6-register window.
    const int w = tid;
    float a0 = tile[0][w + 3], a1 = tile[0][w + 4], a2 = tile[0][w + 5];
    float m0 = tile[1][w + 3], m1 = tile[1][w + 4], m2 = tile[1][w + 5];

    float* op = out + planeOff + (size_t)h0 * WW + w;
#pragma unroll
    for (int ry = 0; ry < TILE_H; ++ry) {
        const float c0 = tile[ry + 2][w + 3];
        const float c1 = tile[ry + 2][w + 4];
        const float c2 = tile[ry + 2][w + 5];
        float s = k0 * a0;
        s = fmaf(k1, a1, s);
        s = fmaf(k2, a2, s);
        s = fmaf(k3, m0, s);
        s = fmaf(k4, m1, s);
        s = fmaf(k5, m2, s);
        s = fmaf(k6, c0, s);
        s = fmaf(k7, c1, s);
        s = fmaf(k8, c2, s);
        op[(size_t)ry * WW] = s;
        a0 = m0; a1 = m1; a2 = m2;
        m0 = c0; m1 = c1; m2 = c2;
    }
}

extern "C" void kernel_launch(void* const* d_in, const int* in_sizes, int n_in,
                              void* d_out, int out_size, void* d_ws, size_t ws_size,
                              hipStream_t stream) {
    const float* x = (const float*)d_in[0];   // 16*64*256*256 f32
    const float* k = (const float*)d_in[1];   // 16*3*3 f32
    float* out = (float*)d_out;               // 16*64*256*256 f32
    (void)in_sizes; (void)n_in; (void)out_size; (void)d_ws; (void)ws_size;

    dim3 grid(BB * CC * TILES_PER_PLANE);     // 8192 tiles
    dim3 block(256);
    dwconv3x3_kernel<<<grid, block, 0, stream>>>(x, k, out);
}